// Net_43533788512497
// MI455X (gfx1250) — compile-verified
//
#include <hip/hip_runtime.h>
#include <math.h>

typedef float v2f __attribute__((ext_vector_type(2)));
typedef float v8f __attribute__((ext_vector_type(8)));

#define NN 100000
#define EE 1600000
#define GG 64
#define CC 8
#define HH 128

// ---------------- zero fill ----------------
__global__ __launch_bounds__(256) void k_zero(float* __restrict__ p, long n) {
  long i = (long)blockIdx.x * blockDim.x + threadIdx.x;
  if (i < n) p[i] = 0.0f;
}

// ---------------- fuse layer-2 weights: Wf[k][col] ----------------
// k<128 -> Wrel2[col][k] ; k>=128 -> Wroot2[col][k-128]
__global__ __launch_bounds__(256) void k_prep_w(const float* __restrict__ Wrel2,
                                                const float* __restrict__ Wroot2,
                                                float* __restrict__ Wf) {
  int i = blockIdx.x * blockDim.x + threadIdx.x;
  if (i >= 256 * 128) return;
  int k = i >> 7, col = i & 127;
  Wf[i] = (k < 128) ? Wrel2[col * 128 + k] : Wroot2[col * 128 + (k - 128)];
}

// ---------------- layer-1 edge scatter: agg1[dst] += x[src][2:5] ----------------
__global__ __launch_bounds__(256) void k_edge1(const float* __restrict__ x,
                                               const int* __restrict__ eidx,
                                               float* __restrict__ agg1) {
  int e = blockIdx.x * blockDim.x + threadIdx.x;
  if (e >= EE) return;
  int s = eidx[e], d = eidx[EE + e];
  const float* xs = x + (long)s * 5 + 2;
  float* ap = agg1 + (long)d * 3;
  atomicAdd(ap + 0, xs[0]);
  atomicAdd(ap + 1, xs[1]);
  atomicAdd(ap + 2, xs[2]);
}

// ---------------- layer-1 node: x1 = relu(agg1@Wrel1^T + xs@Wroot1^T + b) ------
// also fuse segment_max of x1 into g[:,0:128]
__global__ __launch_bounds__(256) void k_node1(const float* __restrict__ x,
                                               const float* __restrict__ agg1,
                                               const float* __restrict__ Wroot1,
                                               const float* __restrict__ Wrel1,
                                               const float* __restrict__ brel1,
                                               const int* __restrict__ batch,
                                               float* __restrict__ x1,
                                               float* __restrict__ g) {
  long idx = (long)blockIdx.x * blockDim.x + threadIdx.x;
  if (idx >= (long)NN * HH) return;
  int node = (int)(idx >> 7);
  int h = (int)(idx & 127);
  float xs0 = x[node * 5 + 2], xs1 = x[node * 5 + 3], xs2 = x[node * 5 + 4];
  float a0 = agg1[node * 3 + 0], a1 = agg1[node * 3 + 1], a2 = agg1[node * 3 + 2];
  float v = brel1[h]
          + Wroot1[h * 3 + 0] * xs0 + Wroot1[h * 3 + 1] * xs1 + Wroot1[h * 3 + 2] * xs2
          + Wrel1[h * 3 + 0] * a0 + Wrel1[h * 3 + 1] * a1 + Wrel1[h * 3 + 2] * a2;
  v = fmaxf(v, 0.0f);
  x1[idx] = v;
  // nonnegative float bits are monotone as int -> int atomicMax == float max
  atomicMax((int*)&g[(long)batch[node] * 256 + h], __float_as_int(v));
}

// ---------------- layer-2 edge scatter: agg2[dst] += x1[src] (128 f32) --------
// one wave per edge: lane handles a float4 chunk
__global__ __launch_bounds__(256) void k_edge2(const float* __restrict__ x1,
                                               const int* __restrict__ eidx,
                                               float* __restrict__ agg2) {
  int e = blockIdx.x * 8 + (threadIdx.x >> 5);
  if (e >= EE) return;
  int lane = threadIdx.x & 31;
  int s = eidx[e], d = eidx[EE + e];
  float4 v = *(const float4*)(x1 + (long)s * 128 + lane * 4);
  float* ap = agg2 + (long)d * 128 + lane * 4;
  atomicAdd(ap + 0, v.x);
  atomicAdd(ap + 1, v.y);
  atomicAdd(ap + 2, v.z);
  atomicAdd(ap + 3, v.w);
}

// ---------------- layer-2 node GEMM on WMMA (f32 16x16x4) ---------------------
// x2 = relu([agg2|x1] @ Wf + brel2), fused directly into segment_max g[:,128:256]
__global__ __launch_bounds__(256) void k_layer2(const float* __restrict__ agg2,
                                                const float* __restrict__ x1,
                                                const float* __restrict__ Wf,
                                                const float* __restrict__ brel2,
                                                const int* __restrict__ batch,
                                                float* __restrict__ g) {
  __shared__ float At[16 * 260];  // 16 rows x 256 cols, stride 260 (bank-decorrelated)
  int row0 = blockIdx.x * 16;
  int tid = threadIdx.x;

  // stage A tile: cols 0..127 <- agg2, 128..255 <- x1
  for (int i = tid; i < 1024; i += 256) {  // 1024 float4 transfers
    int r = i >> 6;
    int c = (i & 63) * 4;
    int node = row0 + r;
    float4 v;
    if (node < NN) {
      if (c < 128) v = *(const float4*)(agg2 + (long)node * 128 + c);
      else         v = *(const float4*)(x1   + (long)node * 128 + (c - 128));
    } else {
      v = make_float4(0.f, 0.f, 0.f, 0.f);
    }
    *(float4*)(&At[r * 260 + c]) = v;
  }
  __syncthreads();

  int wave = tid >> 5;        // 0..7 -> 16-column tile
  int lane = tid & 31;
  int m = lane & 15;          // A row / B,D column index within tile
  int half = lane >> 4;       // K sub-offset select (ISA 16x4 f32 layout)
  int col = wave * 16 + m;    // output column 0..127

  v8f acc = {};
#pragma unroll 8
  for (int k0 = 0; k0 < 256; k0 += 4) {
    int ka = k0 + half * 2;
    // A 16x4: lane holds A[m][ka], A[m][ka+1]
    v2f a = *(const v2f*)(&At[m * 260 + ka]);
    // B 4x16: VGPR0 = B[ka][col], VGPR1 = B[ka+1][col]
    v2f b;
    b.x = Wf[ka * 128 + col];
    b.y = Wf[(ka + 1) * 128 + col];
    acc = __builtin_amdgcn_wmma_f32_16x16x4_f32(
        /*neg_a=*/false, a, /*neg_b=*/false, b,
        /*c_mod=*/(short)0, acc, /*reuse_a=*/false, /*reuse_b=*/false);
  }

  float bias = brel2[col];
#pragma unroll
  for (int j = 0; j < 8; ++j) {
    int node = row0 + j + half * 8;  // D layout: VGPR j holds rows j (lanes 0-15), j+8 (16-31)
    if (node < NN) {
      float v = fmaxf(acc[j] + bias, 0.0f);
      atomicMax((int*)&g[(long)batch[node] * 256 + 128 + col], __float_as_int(v));
    }
  }
}

// ---------------- final MLP + log_softmax (tiny: 64 rows) ---------------------
__global__ __launch_bounds__(64) void k_mlp(const float* __restrict__ g,
                                            const float* __restrict__ exinfo,
                                            const float* __restrict__ W1, const float* __restrict__ b1,
                                            const float* __restrict__ W2, const float* __restrict__ b2,
                                            const float* __restrict__ W3, const float* __restrict__ b3,
                                            float* __restrict__ out) {
  __shared__ float sin_[266];
  __shared__ float s1[64];
  __shared__ float s2[32];
  __shared__ float s3[8];
  int gi = blockIdx.x, tid = threadIdx.x;
  for (int i = tid; i < 256; i += 64) sin_[i] = g[gi * 256 + i];
  if (tid < 10) sin_[256 + tid] = exinfo[gi * 10 + tid];
  __syncthreads();
  {
    float a = b1[tid];
    for (int k = 0; k < 266; ++k) a += W1[tid * 266 + k] * sin_[k];
    s1[tid] = fmaxf(a, 0.0f);
  }
  __syncthreads();
  if (tid < 32) {
    float a = b2[tid];
    for (int k = 0; k < 64; ++k) a += W2[tid * 64 + k] * s1[k];
    s2[tid] = fmaxf(a, 0.0f);
  }
  __syncthreads();
  if (tid < 8) {
    float a = b3[tid];
    for (int k = 0; k < 32; ++k) a += W3[tid * 32 + k] * s2[k];
    s3[tid] = a;
  }
  __syncthreads();
  if (tid == 0) {
    float m = s3[0];
    for (int c = 1; c < 8; ++c) m = fmaxf(m, s3[c]);
    float sum = 0.0f;
    for (int c = 0; c < 8; ++c) sum += expf(s3[c] - m);
    float lse = m + logf(sum);
    for (int c = 0; c < 8; ++c) out[gi * 8 + c] = s3[c] - lse;
  }
}

extern "C" void kernel_launch(void* const* d_in, const int* in_sizes, int n_in,
                              void* d_out, int out_size, void* d_ws, size_t ws_size,
                              hipStream_t stream) {
  const float* x      = (const float*)d_in[0];
  const int*   eidx   = (const int*)d_in[1];
  const int*   batch  = (const int*)d_in[2];
  const float* exinfo = (const float*)d_in[3];
  const float* Wroot1 = (const float*)d_in[4];
  const float* Wrel1  = (const float*)d_in[5];
  const float* brel1  = (const float*)d_in[6];
  const float* Wroot2 = (const float*)d_in[7];
  const float* Wrel2  = (const float*)d_in[8];
  const float* brel2  = (const float*)d_in[9];
  const float* W1     = (const float*)d_in[10];
  const float* b1     = (const float*)d_in[11];
  const float* W2     = (const float*)d_in[12];
  const float* b2     = (const float*)d_in[13];
  const float* W3     = (const float*)d_in[14];
  const float* b3     = (const float*)d_in[15];
  float* out = (float*)d_out;

  float* ws   = (float*)d_ws;
  float* agg1 = ws;                    // 3N
  float* x1   = agg1 + 3L * NN;        // 128N
  float* agg2 = x1 + 128L * NN;        // 128N
  float* gbuf = agg2 + 128L * NN;      // 64*256
  float* Wf   = gbuf + 64L * 256;      // 256*128

  // zero accumulators (every call: graph replay must be deterministic)
  {
    long n1 = 3L * NN;
    long n2 = 128L * NN;
    long n3 = 64L * 256;
    k_zero<<<(int)((n1 + 255) / 256), 256, 0, stream>>>(agg1, n1);
    k_zero<<<(int)((n2 + 255) / 256), 256, 0, stream>>>(agg2, n2);
    k_zero<<<(int)((n3 + 255) / 256), 256, 0, stream>>>(gbuf, n3);
  }
  k_prep_w<<<(256 * 128 + 255) / 256, 256, 0, stream>>>(Wrel2, Wroot2, Wf);

  k_edge1<<<(EE + 255) / 256, 256, 0, stream>>>(x, eidx, agg1);

  {
    long n = (long)NN * HH;
    k_node1<<<(int)((n + 255) / 256), 256, 0, stream>>>(x, agg1, Wroot1, Wrel1, brel1,
                                                        batch, x1, gbuf);
  }

  k_edge2<<<(EE + 7) / 8, 256, 0, stream>>>(x1, eidx, agg2);

  k_layer2<<<(NN + 15) / 16, 256, 0, stream>>>(agg2, x1, Wf, brel2, batch, gbuf);

  k_mlp<<<GG, 64, 0, stream>>>(gbuf, exinfo, W1, b1, W2, b2, W3, b3, out);
}